// HDynamicLayer_83150566851379
// MI455X (gfx1250) — compile-verified
//
#include <hip/hip_runtime.h>

// ---------------------------------------------------------------------------
// h-dynamics sequential scan, CDNA5 (gfx1250, wave32).
//   * one wave per (b,x,y) pixel; h resident in VGPRs (float4 per lane)
//   * weight-row gather: depth-2 async DMA pipeline through a 4-slot LDS ring
//       global_load_async_to_lds_b128 (ASYNCcnt) + s_wait_asynccnt 2
//   * denom: ones-matrix reduction on the matrix pipe
//       2x v_wmma_f32_16x16x4_f32 (exact f32) + 7-add tree + 1 shfl_xor
//     instead of 5 serially-dependent cross-lane shuffles
//   * epilogue: LDS transpose + global_store_async_from_lds_b128 (coalesced)
// ---------------------------------------------------------------------------

typedef float v2f __attribute__((ext_vector_type(2)));
typedef float v8f __attribute__((ext_vector_type(8)));

namespace {
constexpr int kT  = 256;    // timesteps
constexpr int kXY = 256;    // 16*16 pixels per batch element
constexpr int kH  = 128;    // hidden size
constexpr float kEpsGuard = 1e-10f;
}

// Async copy of 16B global -> LDS per lane.
__device__ __forceinline__ void async_row_b128(unsigned lds_off,
                                               unsigned long long gaddr) {
    asm volatile("global_load_async_to_lds_b128 %0, %1, off"
                 :: "v"(lds_off), "v"(gaddr)
                 : "memory");
}

// Async copy of 16B LDS -> global per lane.
__device__ __forceinline__ void async_store_b128(unsigned long long gaddr,
                                                 unsigned lds_off) {
    asm volatile("global_store_async_from_lds_b128 %0, %1, off"
                 :: "v"(gaddr), "v"(lds_off)
                 : "memory");
}

__device__ __forceinline__ void wait_async_le2() {
    asm volatile("s_wait_asynccnt 0x2" ::: "memory");
}

__global__ __launch_bounds__(256, 2) void h_dyn_scan(
    const int*   __restrict__ spikes,   // (B,T,16,16) int32
    const float* __restrict__ eps_xy,   // (16,16,1)
    const float* __restrict__ eps_t,    // (T,)
    const float* __restrict__ weights,  // (N_IN, H)
    const float* __restrict__ h_init,   // (H,)
    float*       __restrict__ out)      // (B,H,16,16)
{
    __shared__ __align__(16) float s_w[8][4][kH];   // 4-slot row ring per wave
    __shared__ float s_et[kT];
    __shared__ int   s_idx[8][kT];
    __shared__ __align__(16) float s_out[kH][8];    // h' x pixel transpose tile

    const int tid  = threadIdx.x;
    const int lane = tid & 31;
    const int wv   = tid >> 5;

    const int b   = blockIdx.x >> 5;        // 32 blocks per batch element
    const int xy0 = (blockIdx.x & 31) * 8;  // first pixel xy of this block
    const int xy  = xy0 + wv;               // this wave's pixel

    // Stage epsilon_t and the block's spike indices (32B-contiguous chunks).
    s_et[tid] = eps_t[tid];
#pragma unroll
    for (int i = 0; i < 8; ++i) {
        const int idx = tid + i * 256;          // 0 .. 2047
        const int t = idx >> 3, j = idx & 7;
        s_idx[j][t] = spikes[(b * kT + t) * kXY + xy0 + j];
    }
    __syncthreads();

    const float eps_p = eps_xy[xy];
    float4 h = *(const float4*)(h_init + lane * 4);   // h[4*lane .. 4*lane+3]

    // Lane's LDS base inside this wave's ring (slot s at +512*s bytes).
    const unsigned lring =
        (unsigned)(unsigned long long)(uintptr_t)&s_w[wv][0][lane * 4];
    const unsigned long long wbase =
        (unsigned long long)(uintptr_t)weights + (unsigned long long)lane * 16ull;

    // All-ones B operand: with B == 1, D[i][j] = sum_k A[i][k] regardless of
    // the B-matrix lane layout.
    const v2f bones = {1.0f, 1.0f};

    // Prologue: start DMA for rows t = 0, 1 (slots 0, 1).
    async_row_b128(lring,       wbase + (unsigned long long)s_idx[wv][0] * 512ull);
    async_row_b128(lring + 512, wbase + (unsigned long long)s_idx[wv][1] * 512ull);

#pragma unroll 4
    for (int t = 0; t < kT; ++t) {
        // Prefetch row t+2 into slot (t+2)&3 (clamped duplicates at the tail
        // keep the "wait<=2 => row t landed" invariant; in-order completion).
        const int tn = (t + 2 < kT) ? (t + 2) : (kT - 1);
        async_row_b128(lring + (unsigned)((t + 2) & 3) * 512u,
                       wbase + (unsigned long long)s_idx[wv][tn] * 512ull);
        wait_async_le2();                        // row t is now in LDS

        const float4 w4 = *(const float4*)&s_w[wv][t & 3][lane * 4];

        float4 hw;
        hw.x = h.x * w4.x;  hw.y = h.y * w4.y;
        hw.z = h.z * w4.z;  hw.w = h.w * w4.w;

        // ---- denom = sum_{128} hw, via ones-matrix WMMA reduction ----
        // A-layout (16x4 f32): VGPR0 = {K0 lanes 0-15 | K2 lanes 16-31},
        // VGPR1 = {K1 | K3}. Packing (x,y) then (z,w) and accumulating gives
        // D[m][*] = 4sum(lane m) + 4sum(lane m+16).
        v2f a0; a0.x = hw.x; a0.y = hw.y;
        v2f a1; a1.x = hw.z; a1.y = hw.w;
        v8f acc = {};
        acc = __builtin_amdgcn_wmma_f32_16x16x4_f32(false, a0, false, bones,
                                                    (short)0, acc, false, false);
        acc = __builtin_amdgcn_wmma_f32_16x16x4_f32(false, a1, false, bones,
                                                    (short)0, acc, false, false);
        // D VGPR v holds row v (lanes 0-15) / row v+8 (lanes 16-31): the 8-reg
        // tree gives rows 0-7 on the low half, rows 8-15 on the high half.
        float s = ((acc[0] + acc[1]) + (acc[2] + acc[3])) +
                  ((acc[4] + acc[5]) + (acc[6] + acc[7]));
        const float denom = s + __shfl_xor(s, 16, 32);

        const float eps = eps_p * s_et[t];
        if (denom > kEpsGuard) {                 // wave-uniform predicate
            const float inv = eps / denom;
            const float r   = 1.0f / (1.0f + eps);
            h.x = (h.x + hw.x * inv) * r;
            h.y = (h.y + hw.y * inv) * r;
            h.z = (h.z + hw.z * inv) * r;
            h.w = (h.w + hw.w * inv) * r;
        }
    }

    // ---- Epilogue: transpose through LDS, then coalesced async DMA out ----
    s_out[4 * lane + 0][wv] = h.x;
    s_out[4 * lane + 1][wv] = h.y;
    s_out[4 * lane + 2][wv] = h.z;
    s_out[4 * lane + 3][wv] = h.w;
    __syncthreads();

    // Thread tid covers one 16B-contiguous global chunk:
    //   out[b][hidx][xy0 + jh .. xy0 + jh + 3]  <-  s_out[hidx][jh .. jh+3]
    const int hidx = tid >> 1;
    const int jh   = (tid & 1) * 4;
    const unsigned lsrc =
        (unsigned)(unsigned long long)(uintptr_t)&s_out[hidx][jh];
    const unsigned long long gdst = (unsigned long long)(uintptr_t)
        (out + (size_t)(b * kH + hidx) * kXY + xy0 + jh);
    async_store_b128(gdst, lsrc);
    // s_endpgm performs an implicit wait-idle, covering the pending ASYNCcnt.
}

extern "C" void kernel_launch(void* const* d_in, const int* in_sizes, int n_in,
                              void* d_out, int out_size, void* d_ws, size_t ws_size,
                              hipStream_t stream) {
    // setup_inputs order:
    // 0:input 1:spikes 2:epsilon_xy 3:epsilon_t_0 4:weights 5:h_initial
    // 6:last_grad_scale 7:labels   (0,6,7 unused by the reference)
    const int*   spikes  = (const int*)  d_in[1];
    const float* eps_xy  = (const float*)d_in[2];
    const float* eps_t   = (const float*)d_in[3];
    const float* weights = (const float*)d_in[4];
    const float* h_init  = (const float*)d_in[5];
    float*       out     = (float*)d_out;

    // 2048 pixels / 8 waves per block = 256 blocks of 256 threads.
    h_dyn_scan<<<256, 256, 0, stream>>>(spikes, eps_xy, eps_t, weights,
                                        h_init, out);
}